// MultiLayerRNN_85272280695138
// MI455X (gfx1250) — compile-verified
//
#include <hip/hip_runtime.h>
#include <hip/hip_bf16.h>

// MI455X (gfx1250) multi-layer RNN forward.
// Memory-bound fp32 GEMV chain; dot products via V_WMMA_F32_16X16X4_F32.
// Whh is mathematically dead (hidden state is zero), so it is never read.
// Traffic: 9 x 64 MiB fp32 weights, read once -> ~26 us floor at 23.3 TB/s.

#define HDIM 4096
#define WAVES_PER_WG 8
#define THREADS (WAVES_PER_WG * 32)
#define KSEG (HDIM / WAVES_PER_WG)   // 512 K-elements per wave

typedef __attribute__((ext_vector_type(2))) float v2f;
typedef __attribute__((ext_vector_type(8))) float v8f;

__global__ __launch_bounds__(THREADS)
void rnn_gemv_wmma(const float* __restrict__ x,    // [HDIM] input vector
                   const float* __restrict__ W,    // [HDIM, HDIM] row-major; out[n]=dot(x, W[n,:])
                   const float* __restrict__ b1,   // [HDIM]
                   const float* __restrict__ b2,   // [HDIM] or nullptr
                   float* __restrict__ out,        // [HDIM]
                   int applyTanh) {
    __shared__ __align__(16) float xs[HDIM];
    __shared__ float partials[WAVES_PER_WG * 16];

    const int tid  = threadIdx.x;
    const int lane = tid & 31;
    const int wave = tid >> 5;
    const int nbase = blockIdx.x * 16;

    // Cooperatively stage x into LDS (16 KB), float4-wide, branchless
    // (constant trip count so no EXEC-mask ladder).
    {
        const float4* x4  = reinterpret_cast<const float4*>(x);
        float4*       xs4 = reinterpret_cast<float4*>(xs);
        #pragma unroll
        for (int j = 0; j < HDIM / 4 / THREADS; ++j)
            xs4[tid + j * THREADS] = x4[tid + j * THREADS];
    }
    __syncthreads();

    // WMMA operand addressing per ISA 16x4 f32 A-layout:
    //   lane l (0-15):  A[l][0..1]    -> W[row][k+0], W[row][k+1]
    //   lane l (16-31): A[l-16][2..3] -> W[row][k+2], W[row][k+3]
    // B (4x16) mirrors this with N striped over lanes; we broadcast x into all
    // 16 columns so every column of D holds the same dot product.
    const int  row  = nbase + (lane & 15);
    const int  koff = (lane >> 4) << 1;          // 0 or 2
    const float* wrow = W + (size_t)row * HDIM + koff;

    // Two independent accumulator chains so back-to-back v_wmma ops don't
    // serialize on the D->C dependency; loads interleave freely.
    v8f acc0 = {};
    v8f acc1 = {};
    const int k0   = wave * KSEG;
    const int kend = k0 + KSEG;

    #pragma unroll 2
    for (int k = k0; k < kend; k += 8) {
        v2f a0 = *reinterpret_cast<const v2f*>(wrow + k);           // global_load_b64 (W)
        v2f b0 = *reinterpret_cast<const v2f*>(&xs[k + koff]);      // ds_load_b64 (x bcast)
        v2f a1 = *reinterpret_cast<const v2f*>(wrow + k + 4);
        v2f b1v = *reinterpret_cast<const v2f*>(&xs[k + 4 + koff]);
        acc0 = __builtin_amdgcn_wmma_f32_16x16x4_f32(
                   false, a0, false, b0, (short)0, acc0, false, false);
        acc1 = __builtin_amdgcn_wmma_f32_16x16x4_f32(
                   false, a1, false, b1v, (short)0, acc1, false, false);
    }
    v8f acc = acc0 + acc1;

    // D layout: VGPR r -> lanes 0-15 hold M=r, lanes 16-31 hold M=r+8
    // (all columns equal, so lanes 0 and 16 carry the 16 tile outputs).
    if (lane == 0) {
        #pragma unroll
        for (int r = 0; r < 8; ++r) partials[wave * 16 + r] = acc[r];
    } else if (lane == 16) {
        #pragma unroll
        for (int r = 0; r < 8; ++r) partials[wave * 16 + 8 + r] = acc[r];
    }
    __syncthreads();

    // Cross-wave K reduction + bias + activation for this block's 16 outputs.
    if (tid < 16) {
        const int n = nbase + tid;
        float s = 0.0f;
        #pragma unroll
        for (int w = 0; w < WAVES_PER_WG; ++w) s += partials[w * 16 + tid];
        s += b1[n];
        if (b2) s += b2[n];
        out[n] = applyTanh ? tanhf(s) : s;
    }
}

extern "C" void kernel_launch(void* const* d_in, const int* in_sizes, int n_in,
                              void* d_out, int out_size, void* d_ws, size_t ws_size,
                              hipStream_t stream) {
    (void)in_sizes; (void)n_in; (void)out_size; (void)ws_size;

    const float* x    = (const float*)d_in[0];  // [1, H]
    const float* Wxh  = (const float*)d_in[1];  // [8, H, H]
    const float* bxh  = (const float*)d_in[2];  // [8, H]
    // d_in[3] = Whh: multiplied by a zero hidden state -> never read.
    const float* bhh  = (const float*)d_in[4];  // [8, H]
    const float* fc_w = (const float*)d_in[5];  // [H, H]
    const float* fc_b = (const float*)d_in[6];  // [H]
    float* out = (float*)d_out;

    float* h0 = (float*)d_ws;                   // ping-pong hidden buffers
    float* h1 = h0 + HDIM;

    const dim3 grid(HDIM / 16);                 // 256 tiles of 16 outputs
    const dim3 block(THREADS);                  // 8 wave32

    const float* cur = x;
    float* nxt = h0;
    for (int i = 0; i < 8; ++i) {
        rnn_gemv_wmma<<<grid, block, 0, stream>>>(
            cur, Wxh + (size_t)i * HDIM * HDIM, bxh + (size_t)i * HDIM,
            bhh + (size_t)i * HDIM, nxt, /*applyTanh=*/1);
        cur = nxt;
        nxt = (nxt == h0) ? h1 : h0;
    }
    // Final FC: single bias, no tanh, writes d_out.
    rnn_gemv_wmma<<<grid, block, 0, stream>>>(
        cur, fc_w, fc_b, /*b2=*/nullptr, out, /*applyTanh=*/0);
}